// Net_55860344652158
// MI455X (gfx1250) — compile-verified
//
#include <hip/hip_runtime.h>
#include <cstdint>

typedef __attribute__((ext_vector_type(2))) float v2f;
typedef __attribute__((ext_vector_type(8))) float v8f;
typedef __attribute__((ext_vector_type(4))) int   v4i;

#define SNN_T    100
#define SNN_NIN  784
#define SNN_NH   100
#define SNN_NOUT 10
#define SNN_B    1024
#define NKQ      196            // K quads per step (784/4)
#define XSTR     788            // padded LDS row stride (floats) -> conflict-free A reads
#define XBUF     (16*XSTR)      // floats per x tile buffer
#define S2STR    116            // padded stride for staged s2 accumulator
#define NTHREADS 224            // 7 waves: one per 16-wide hidden tile

#if defined(__HIP_DEVICE_COMPILE__) && __has_builtin(__builtin_amdgcn_global_load_async_to_lds_b128)
#define HAVE_ASYNC 1
#else
#define HAVE_ASYNC 0
#endif

__device__ __forceinline__ void copy16(const float* g, float* l) {
#if HAVE_ASYNC
    __builtin_amdgcn_global_load_async_to_lds_b128(
        (__attribute__((address_space(1))) v4i*)(g),
        (__attribute__((address_space(3))) v4i*)(l), 0, 0);
#else
    *(float4*)l = *(const float4*)g;
#endif
}

__device__ __forceinline__ void wait_async14() {
#if HAVE_ASYNC
#if __has_builtin(__builtin_amdgcn_s_wait_asynccnt)
    __builtin_amdgcn_s_wait_asynccnt(14);
#else
    asm volatile("s_wait_asynccnt 14" ::: "memory");
#endif
#endif
}

__device__ __forceinline__ void wait_async0() {
#if HAVE_ASYNC
#if __has_builtin(__builtin_amdgcn_s_wait_asynccnt)
    __builtin_amdgcn_s_wait_asynccnt(0);
#else
    asm volatile("s_wait_asynccnt 0" ::: "memory");
#endif
#endif
}

__global__ __launch_bounds__(NTHREADS) void snn_fused_kernel(
    const float* __restrict__ x,  const float* __restrict__ pws1,
    const float* __restrict__ W1, const float* __restrict__ b1,
    const float* __restrict__ pws2, const float* __restrict__ W2,
    const float* __restrict__ b2, float* __restrict__ out)
{
    extern __shared__ float smem[];          // 2*XBUF x-tile buffers + s2 staging
    const int tid  = threadIdx.x;
    const int wid  = tid >> 5;               // wave id 0..6 = hidden tile
    const int lane = tid & 31;
    const int Lm   = lane & 15;
    const int hi   = lane >> 4;              // lane-half selects K-pair (A/B) / M-half (C)
    const int b0   = blockIdx.x << 4;        // 16 batch rows per WG

    float* xs  = smem;
    float* s2m = smem + 2*XBUF;

    const float inv1 = 1.0f / (1.0f + __expf(-pws1[0]));
    const float inv2 = 1.0f / (1.0f + __expf(-pws2[0]));
    const float d1 = 1.0f - inv1;
    const float d2 = 1.0f - inv2;

    // This wave's B-fragment source: hidden unit col = 16*wid + Lm.
    // Padded cols (>=100) alias W1 row 0: they accumulate finite garbage that is
    // annihilated in GEMM2 (W2 fragment is zeroed for kb >= 100).
    const int   col   = (wid << 4) + Lm;
    const float* w1row = W1 + (size_t)((col < SNN_NH) ? col : 0) * SNN_NIN + 2*hi;
    const float b1v   = (col < SNN_NH) ? b1[col] : 0.0f;

    // W1 is time-invariant: preload this wave's whole B tile into registers
    // (196 v2f = 392 VGPRs; CDNA5 wave32 supports up to 1024 VGPRs/wave via
    // compiler-managed VGPR-MSB windows). Removes all global loads from the
    // 19600-WMMA hot loop.
    v2f wreg[NKQ];
    #pragma unroll
    for (int k = 0; k < NKQ; ++k)
        wreg[k] = *(const v2f*)(w1row + (k << 2));

    // Persistent LIF/filter state, kept in WMMA C layout (row M = r + 8*hi, col N = Lm)
    v8f h1t = {}, v1 = {}, s2 = {}, s2a = {};

    // Prologue: async-stage x tile for t=0 into buffer 0 (3136 b128 chunks / 224 lanes = 14 each)
    #pragma unroll
    for (int i = 0; i < 14; ++i) {
        int idx = i*NTHREADS + tid;
        int row = idx / 196;
        int c4  = (idx - row*196) << 2;
        copy16(x + ((size_t)(b0+row)*SNN_T + 0)*SNN_NIN + c4, xs + row*XSTR + c4);
    }

    int cur = 0;
    for (int t = 0; t < SNN_T; ++t) {
        if (t + 1 < SNN_T) {                 // prefetch next tile into the other buffer
            float* dst = xs + (cur ^ 1)*XBUF;
            #pragma unroll
            for (int i = 0; i < 14; ++i) {
                int idx = i*NTHREADS + tid;
                int row = idx / 196;
                int c4  = (idx - row*196) << 2;
                copy16(x + ((size_t)(b0+row)*SNN_T + (t+1))*SNN_NIN + c4, dst + row*XSTR + c4);
            }
            wait_async14();                  // older (current-tile) batch complete
        } else {
            wait_async0();
        }
        __syncthreads();

        // GEMM1: u = Xtile(16x784) * W1_tile^T(784x16); A from LDS, B from registers
        const float* xr = xs + cur*XBUF + Lm*XSTR + 2*hi;
        v8f acc = {};
        #pragma unroll
        for (int kq = 0; kq < NKQ; ++kq) {
            v2f a = *(const v2f*)(xr + (kq << 2));   // A: rows=batch, K pair per lane half
            acc = __builtin_amdgcn_wmma_f32_16x16x4_f32(
                false, a, false, wreg[kq], (short)0, acc, false, false);
        }

        // Fused filter + LIF1 + synapse2 + weighted v2 accumulator (all in C layout)
        #pragma unroll
        for (int r = 0; r < 8; ++r) {
            float h = h1t[r]*d1 + acc[r];            // h1~ = decay*h1~ + x_t@W1^T
            h1t[r] = h;
            float v = 0.5f*(v1[r] + h + b1v);        // v += (h1 - v)/tau, tau=2
            float sp = (v >= 1.0f) ? 1.0f : 0.0f;    // Heaviside(v - 1)
            v1[r] = (v >= 1.0f) ? 0.0f : v;          // hard reset
            float s = s2[r]*d2 + sp;                 // synapse filter 2
            s2[r] = s;
            s2a[r] = 0.5f*(s2a[r] + s);              // A_t = (A_{t-1}+s2_t)/2  (== v2 scan)
        }
        __syncthreads();                             // protect buffer before next prefetch
        cur ^= 1;
    }

    // Stage weighted synapse accumulator as a 16x112 matrix in LDS
    #pragma unroll
    for (int r = 0; r < 8; ++r) {
        int row = r + (hi << 3);
        s2m[row*S2STR + (wid << 4) + Lm] = s2a[r];
    }
    __syncthreads();

    // GEMM2 (hoisted out of the time loop): v2 = A @ W2^T + b2  -> wave 0, 28 WMMAs
    if (wid == 0) {
        v8f c2 = {};
        const int oc = Lm;                           // output channel (pad >=10)
        const int ok = (oc < SNN_NOUT) ? 1 : 0;
        #pragma unroll
        for (int k0 = 0; k0 < 112; k0 += 4) {
            int kb = k0 + 2*hi;
            v2f a = { s2m[Lm*S2STR + kb], s2m[Lm*S2STR + kb + 1] };
            float e0 = (ok && kb     < SNN_NH) ? W2[oc*SNN_NH + kb    ] : 0.0f;
            float e1 = (ok && kb + 1 < SNN_NH) ? W2[oc*SNN_NH + kb + 1] : 0.0f;
            v2f b = { e0, e1 };
            c2 = __builtin_amdgcn_wmma_f32_16x16x4_f32(
                false, a, false, b, (short)0, c2, false, false);
        }
        if (oc < SNN_NOUT) {
            float b2v = b2[oc];                      // *(1 - 2^-100) == 1.0f in fp32
            #pragma unroll
            for (int r = 0; r < 8; ++r) {
                int row = r + (hi << 3);
                out[(size_t)(b0 + row)*SNN_NOUT + oc] = c2[r] + b2v;
            }
        }
    }
}

extern "C" void kernel_launch(void* const* d_in, const int* in_sizes, int n_in,
                              void* d_out, int out_size, void* d_ws, size_t ws_size,
                              hipStream_t stream) {
    (void)in_sizes; (void)n_in; (void)d_ws; (void)ws_size; (void)out_size;
    const float* x   = (const float*)d_in[0];
    const float* ws1 = (const float*)d_in[1];
    const float* W1  = (const float*)d_in[2];
    const float* b1  = (const float*)d_in[3];
    const float* ws2 = (const float*)d_in[4];
    const float* W2  = (const float*)d_in[5];
    const float* b2  = (const float*)d_in[6];
    float* out = (float*)d_out;

    const size_t smem_bytes = (size_t)(2*XBUF + 16*S2STR) * sizeof(float); // ~108 KB (<320 KB/WGP)
    snn_fused_kernel<<<dim3(SNN_B/16), dim3(NTHREADS), smem_bytes, stream>>>(
        x, ws1, W1, b1, ws2, W2, b2, out);
}